// GraphSAGEModel_82712480186688
// MI455X (gfx1250) — compile-verified
//
#include <hip/hip_runtime.h>

typedef float v2f __attribute__((ext_vector_type(2)));
typedef float v8f __attribute__((ext_vector_type(8)));

// ---------------- elementwise / setup kernels ----------------

__global__ void zero_f32(float* __restrict__ p, long n) {
  long i = (long)blockIdx.x * blockDim.x + threadIdx.x;
  if (i < n) p[i] = 0.0f;
}

__global__ void degree_kernel(const long long* __restrict__ dst, int E,
                              float* __restrict__ deg) {
  int e = blockIdx.x * blockDim.x + threadIdx.x;
  if (e < E) atomicAdd(&deg[(int)dst[e]], 1.0f);
}

// in-place: deg -> dis = rsqrt(deg + 1)  (self-loop adds 1; result >= 1 so max() is moot)
__global__ void dis_kernel(float* __restrict__ d, int n) {
  int i = blockIdx.x * blockDim.x + threadIdx.x;
  if (i < n) d[i] = rsqrtf(d[i] + 1.0f);
}

// one-time edge preprocessing: int64 -> int32 indices + fused edge norm
__global__ void edge_prep(const long long* __restrict__ src64,
                          const long long* __restrict__ dst64,
                          const float* __restrict__ dis,
                          int* __restrict__ src32, int* __restrict__ dst32,
                          float* __restrict__ norm, int E) {
  int e = blockIdx.x * blockDim.x + threadIdx.x;
  if (e < E) {
    int s = (int)src64[e];
    int d = (int)dst64[e];
    src32[e] = s;
    dst32[e] = d;
    norm[e] = dis[s] * dis[d];
  }
}

// ---------------- WMMA f32 GEMM:  Out[M x HOUT] = A[M x K] @ W[K x HOUT] ----------------
// 32 rows per block (2 row tiles -> each B fragment feeds 2 WMMAs);
// one wave per 16-col tile (blockDim.x = (HOUT/16)*32); K-loop fully unrolled.
template <int K, int HOUT>
__global__ void gemm_wmma_f32(const float* __restrict__ A, const float* __restrict__ W,
                              float* __restrict__ Out) {
  constexpr int STRIDE = K + 4;  // bank-conflict-free padding
  __shared__ float As[32 * STRIDE];
  const int rowBase = blockIdx.x * 32;

  // cooperative, coalesced load of the 32 x K A-tile into LDS
  for (int idx = threadIdx.x; idx < 32 * K; idx += blockDim.x) {
    int r = idx / K, c = idx - r * K;
    As[r * STRIDE + c] = A[(size_t)(rowBase + r) * K + c];
  }
  __syncthreads();

  const int lane = threadIdx.x & 31;
  const int wave = threadIdx.x >> 5;  // column tile index
  const int m    = lane & 15;         // A row / B col within tile
  const int kb   = (lane >> 4) * 2;   // K sub-offset per half-wave
  const int col  = wave * 16 + m;
  const float* __restrict__ As0 = &As[m * STRIDE];
  const float* __restrict__ As1 = &As[(m + 16) * STRIDE];

  v8f acc0 = {}, acc1 = {};
#pragma unroll
  for (int k = 0; k < K; k += 4) {
    v2f b, a0, a1;
    b.x  = W[(size_t)(k + kb) * HOUT + col];
    b.y  = W[(size_t)(k + kb + 1) * HOUT + col];
    a0.x = As0[k + kb];
    a0.y = As0[k + kb + 1];
    a1.x = As1[k + kb];
    a1.y = As1[k + kb + 1];
    acc0 = __builtin_amdgcn_wmma_f32_16x16x4_f32(false, a0, false, b, (short)0, acc0, false, false);
    acc1 = __builtin_amdgcn_wmma_f32_16x16x4_f32(false, a1, false, b, (short)0, acc1, false, false);
  }

  // C/D layout: VGPR r -> row = r + 8*(lane>=16), col = lane&15
  const int rowOff = 8 * (lane >> 4);
  float* outp0 = Out + (size_t)rowBase * HOUT + wave * 16 + m;
  float* outp1 = outp0 + (size_t)16 * HOUT;
#pragma unroll
  for (int r = 0; r < 8; ++r) {
    outp0[(size_t)(r + rowOff) * HOUT] = acc0[r];
    outp1[(size_t)(r + rowOff) * HOUT] = acc1[r];
  }
}

// ---------------- edge scatter: one wave per edge, vectorized f32 atomics ----------------
template <int H>
__global__ void scatter_kernel(const int* __restrict__ src, const int* __restrict__ dst,
                               const float* __restrict__ norm,
                               const float* __restrict__ h, float* __restrict__ agg, int E) {
  constexpr int VEC = H / 32;
  long t = (long)blockIdx.x * blockDim.x + threadIdx.x;
  long e = t >> 5;
  if (e >= (long)E) return;
  int lane = (int)(t & 31);
  int s = src[e];
  int d = dst[e];
  float nm = norm[e];
  const float* hp = h + (size_t)s * H + lane * VEC;
  float* ap = agg + (size_t)d * H + lane * VEC;
  if (VEC == 4) {
    float4 v = *reinterpret_cast<const float4*>(hp);
    atomicAdd(ap + 0, v.x * nm);
    atomicAdd(ap + 1, v.y * nm);
    atomicAdd(ap + 2, v.z * nm);
    atomicAdd(ap + 3, v.w * nm);
  } else {
    float2 v = *reinterpret_cast<const float2*>(hp);
    atomicAdd(ap + 0, v.x * nm);
    atomicAdd(ap + 1, v.y * nm);
  }
}

// ---------------- fused self-loop + bias + ReLU (in place on agg) ----------------
__global__ void bias_relu_selfloop(float* __restrict__ agg, const float* __restrict__ h,
                                   const float* __restrict__ dis, const float* __restrict__ b,
                                   int n, int H) {
  long t = (long)blockIdx.x * blockDim.x + threadIdx.x;
  if (t >= (long)n * H) return;
  int i = (int)(t / H);
  int j = (int)(t - (long)i * H);
  float di = dis[i];
  float v = agg[t] + h[t] * di * di + b[j];
  agg[t] = v > 0.0f ? v : 0.0f;
}

// ---------------- column-sum pooling over N x 64 ----------------
__global__ void pool_kernel(const float* __restrict__ h3, int n, float* __restrict__ pooled) {
  __shared__ float sh[256];
  int col  = threadIdx.x & 63;
  int rgrp = threadIdx.x >> 6;  // 0..3
  float acc = 0.0f;
  for (long r = (long)blockIdx.x * 4 + rgrp; r < (long)n; r += (long)gridDim.x * 4)
    acc += h3[r * 64 + col];
  sh[threadIdx.x] = acc;
  __syncthreads();
  if (threadIdx.x < 64) {
    float s = sh[threadIdx.x] + sh[threadIdx.x + 64] + sh[threadIdx.x + 128] + sh[threadIdx.x + 192];
    atomicAdd(&pooled[threadIdx.x], s);
  }
}

// ---------------- tiny MLP head + softmax (single block) ----------------
__global__ void head_kernel(const float* __restrict__ pooled,
                            const float* __restrict__ Wc1, const float* __restrict__ bc1,
                            const float* __restrict__ Wc2, const float* __restrict__ bc2,
                            float* __restrict__ out, int n) {
  __shared__ float pm[64];
  __shared__ float z[32];
  int t = threadIdx.x;  // blockDim = 64
  pm[t] = pooled[t] * (1.0f / (float)n);
  __syncthreads();
  if (t < 32) {
    float acc = bc1[t];
    for (int j = 0; j < 64; ++j) acc += pm[j] * Wc1[j * 32 + t];
    z[t] = acc > 0.0f ? acc : 0.0f;
  }
  __syncthreads();
  if (t == 0) {
    float l0 = bc2[0], l1 = bc2[1];
    for (int k = 0; k < 32; ++k) {
      l0 += z[k] * Wc2[k * 2 + 0];
      l1 += z[k] * Wc2[k * 2 + 1];
    }
    float m = fmaxf(l0, l1);
    float e0 = expf(l0 - m), e1 = expf(l1 - m);
    float inv = 1.0f / (e0 + e1);
    out[0] = e0 * inv;
    out[1] = e1 * inv;
  }
}

// ---------------- launch ----------------

extern "C" void kernel_launch(void* const* d_in, const int* in_sizes, int n_in,
                              void* d_out, int out_size, void* d_ws, size_t ws_size,
                              hipStream_t stream) {
  const float*     x   = (const float*)d_in[0];
  const long long* ei  = (const long long*)d_in[1];  // int64 [2, E]
  const float*     W1  = (const float*)d_in[2];
  const float*     b1  = (const float*)d_in[3];
  const float*     W2  = (const float*)d_in[4];
  const float*     b2  = (const float*)d_in[5];
  const float*     W3  = (const float*)d_in[6];
  const float*     b3  = (const float*)d_in[7];
  const float*     Wc1 = (const float*)d_in[8];
  const float*     bc1 = (const float*)d_in[9];
  const float*     Wc2 = (const float*)d_in[10];
  const float*     bc2 = (const float*)d_in[11];

  const int N = 100000, E = 1600000;
  const long long* src64 = ei;
  const long long* dst64 = ei + E;

  char* ws = (char*)d_ws;
  auto alignUp = [](size_t v) { return (v + 255) & ~(size_t)255; };
  size_t off = 0;
  float* dis    = (float*)(ws + off); off += alignUp((size_t)N * 4);       // deg -> dis in place
  float* pooled = (float*)(ws + off); off += 256;                          // 64 floats
  int*   src32  = (int*)(ws + off);   off += alignUp((size_t)E * 4);
  int*   dst32  = (int*)(ws + off);   off += alignUp((size_t)E * 4);
  float* enorm  = (float*)(ws + off); off += alignUp((size_t)E * 4);
  float* buf0   = (float*)(ws + off); off += alignUp((size_t)N * 128 * 4); // agg / activations
  float* buf1   = (float*)(ws + off);                                      // GEMM output h

  const unsigned TB = 256;
  auto blocks = [&](long n) { return (unsigned)((n + TB - 1) / TB); };

  // degrees -> dis -> packed edge data (per-call re-init: deterministic)
  zero_f32<<<blocks(N), TB, 0, stream>>>(dis, N);
  zero_f32<<<1, 64, 0, stream>>>(pooled, 64);
  degree_kernel<<<blocks(E), TB, 0, stream>>>(dst64, E, dis);
  dis_kernel<<<blocks(N), TB, 0, stream>>>(dis, N);
  edge_prep<<<blocks(E), TB, 0, stream>>>(src64, dst64, dis, src32, dst32, enorm, E);

  const long scatterThreads = (long)E * 32;

  // ---- Layer 1: x[N,64] @ W1[64,128] ----
  gemm_wmma_f32<64, 128><<<N / 32, 8 * 32, 0, stream>>>(x, W1, buf1);
  zero_f32<<<blocks((long)N * 128), TB, 0, stream>>>(buf0, (long)N * 128);
  scatter_kernel<128><<<blocks(scatterThreads), TB, 0, stream>>>(src32, dst32, enorm, buf1, buf0, E);
  bias_relu_selfloop<<<blocks((long)N * 128), TB, 0, stream>>>(buf0, buf1, dis, b1, N, 128);

  // ---- Layer 2: h1[N,128] @ W2[128,128] ----
  gemm_wmma_f32<128, 128><<<N / 32, 8 * 32, 0, stream>>>(buf0, W2, buf1);
  zero_f32<<<blocks((long)N * 128), TB, 0, stream>>>(buf0, (long)N * 128);
  scatter_kernel<128><<<blocks(scatterThreads), TB, 0, stream>>>(src32, dst32, enorm, buf1, buf0, E);
  bias_relu_selfloop<<<blocks((long)N * 128), TB, 0, stream>>>(buf0, buf1, dis, b2, N, 128);

  // ---- Layer 3: h2[N,128] @ W3[128,64] ----
  gemm_wmma_f32<128, 64><<<N / 32, 4 * 32, 0, stream>>>(buf0, W3, buf1);
  zero_f32<<<blocks((long)N * 64), TB, 0, stream>>>(buf0, (long)N * 64);
  scatter_kernel<64><<<blocks(scatterThreads), TB, 0, stream>>>(src32, dst32, enorm, buf1, buf0, E);
  bias_relu_selfloop<<<blocks((long)N * 64), TB, 0, stream>>>(buf0, buf1, dis, b3, N, 64);

  // ---- mean pool + MLP head + softmax ----
  pool_kernel<<<1024, 256, 0, stream>>>(buf0, N, pooled);
  head_kernel<<<1, 64, 0, stream>>>(pooled, Wc1, bc1, Wc2, bc2, (float*)d_out, N);
}